// FMoELinear_49804440764685
// MI455X (gfx1250) — compile-verified
//
#include <hip/hip_runtime.h>
#include <hip/hip_bf16.h>

#define NUM_EXPERT 16
#define IN_FEAT    2048
#define OUT_FEAT   8192
#define TOTAL_TOKENS 8192

#define TM 128              // token rows per block tile
#define TN 128              // output features per block tile
#define KB 16               // K chunk staged in LDS
#define KBPAD 20            // padded row stride (words): conflict-free b64 frags, 16B-aligned b128 async stores
#define KCHUNKS (IN_FEAT / KB)
#define MAX_TILES 80        // sum ceil(n_e/TM) <= TOTAL/TM + NUM_EXPERT = 64+16

typedef float v2f __attribute__((ext_vector_type(2)));
typedef float v8f __attribute__((ext_vector_type(8)));

// generic (flat) shared pointer -> 32-bit LDS byte offset for async-to-LDS vdst
__device__ __forceinline__ unsigned lds_off(const float* p) {
    return (unsigned)(unsigned long long)(const __attribute__((address_space(3))) float*)p;
}

__device__ __forceinline__ void wait_async0() {
#if __has_builtin(__builtin_amdgcn_s_wait_asynccnt)
    __builtin_amdgcn_s_wait_asynccnt(0);
#else
    asm volatile("s_wait_asynccnt 0x0" ::: "memory");
#endif
}

// ---- tiny setup kernel: build row-tile table from per-expert counts --------
__global__ void fmoe_build_tiles(const int* __restrict__ counts, int* __restrict__ ws) {
    if (threadIdx.x != 0 || blockIdx.x != 0) return;
    int4* ent = (int4*)(ws + 4);   // entries start 16B into ws
    int off = 0, t = 0;
    for (int e = 0; e < NUM_EXPERT; ++e) {
        int n = counts[e];
        for (int s = 0; s < n; s += TM) {
            int r = n - s; if (r > TM) r = TM;
            ent[t++] = make_int4(e, off + s, r, 0);
        }
        off += n;
    }
    ws[0] = t;
}

// ---- main grouped-GEMM kernel: fp32 WMMA 16x16x4, async global->LDS --------
__global__ __launch_bounds__(256, 1)
void fmoe_linear_wmma(const float* __restrict__ inp,
                      const float* __restrict__ weight,
                      float* __restrict__ out,
                      const int* __restrict__ ws) {
    __shared__ __align__(16) float sA[2][TM * KBPAD];
    __shared__ __align__(16) float sB[2][TN * KBPAD];

    const int numTiles = ws[0];
    if ((int)blockIdx.x >= numTiles) return;               // uniform per block
    const int4 te = ((const int4*)(ws + 4))[blockIdx.x];
    const int expert    = te.x;
    const int row_start = te.y;
    const int rows      = te.z;
    const int obase     = blockIdx.y * TN;

    const int tid  = threadIdx.x;
    const int lane = tid & 31;
    const int wv   = tid >> 5;           // 8 waves
    const int wm   = wv & 3;             // 4 waves along M (32 rows each)
    const int wn   = wv >> 2;            // 2 waves along N (64 cols each)
    const int l16  = lane & 15;
    const int hf   = lane >> 4;

    const float* Abase = inp    + (size_t)row_start * IN_FEAT;
    const float* Bbase = weight + (size_t)expert * OUT_FEAT * IN_FEAT
                                + (size_t)obase * IN_FEAT;

    // per-thread copy slots: 2 float4 of A + 2 of B per chunk (128x16 tiles)
    const int i0   = tid;            // slot 0
    const int i1   = tid + 256;      // slot 1
    const int row0 = i0 >> 2, c40 = (i0 & 3) * 4;
    const int row1 = i1 >> 2, c41 = (i1 & 3) * 4;
    const int arow0 = (row0 < rows) ? row0 : (rows - 1);   // clamp: stay in-segment
    const int arow1 = (row1 < rows) ? row1 : (rows - 1);

    v8f acc[2][4];
    {
        v8f z = {};
#pragma unroll
        for (int mi = 0; mi < 2; ++mi)
#pragma unroll
            for (int ni = 0; ni < 4; ++ni) acc[mi][ni] = z;
    }

    auto fetch_async = [&](int kc, int buf) {
        const int k0 = kc * KB;
        const float* ga0 = Abase + (size_t)arow0 * IN_FEAT + k0 + c40;
        const float* ga1 = Abase + (size_t)arow1 * IN_FEAT + k0 + c41;
        const float* gb0 = Bbase + (size_t)row0  * IN_FEAT + k0 + c40;
        const float* gb1 = Bbase + (size_t)row1  * IN_FEAT + k0 + c41;
        const unsigned la0 = lds_off(&sA[buf][row0 * KBPAD + c40]);
        const unsigned la1 = lds_off(&sA[buf][row1 * KBPAD + c41]);
        const unsigned lb0 = lds_off(&sB[buf][row0 * KBPAD + c40]);
        const unsigned lb1 = lds_off(&sB[buf][row1 * KBPAD + c41]);
        asm volatile("global_load_async_to_lds_b128 %0, %1, off"
                     :: "v"(la0), "v"(ga0) : "memory");
        asm volatile("global_load_async_to_lds_b128 %0, %1, off"
                     :: "v"(la1), "v"(ga1) : "memory");
        asm volatile("global_load_async_to_lds_b128 %0, %1, off"
                     :: "v"(lb0), "v"(gb0) : "memory");
        asm volatile("global_load_async_to_lds_b128 %0, %1, off"
                     :: "v"(lb1), "v"(gb1) : "memory");
    };

    auto compute = [&](int buf) {
        const float* A = sA[buf];
        const float* B = sB[buf];
#pragma unroll
        for (int j = 0; j < 4; ++j) {          // 4 K-steps of 4 inside the chunk
            v2f af[2], bf[4];
#pragma unroll
            for (int mi = 0; mi < 2; ++mi)
                af[mi] = *(const v2f*)&A[(wm * 32 + mi * 16 + l16) * KBPAD + j * 4 + 2 * hf];
#pragma unroll
            for (int ni = 0; ni < 4; ++ni)
                bf[ni] = *(const v2f*)&B[(wn * 64 + ni * 16 + l16) * KBPAD + j * 4 + 2 * hf];
#pragma unroll
            for (int mi = 0; mi < 2; ++mi)
#pragma unroll
                for (int ni = 0; ni < 4; ++ni)
                    acc[mi][ni] = __builtin_amdgcn_wmma_f32_16x16x4_f32(
                        false, af[mi], false, bf[ni],
                        (short)0, acc[mi][ni], false, false);
        }
    };

    // async double-buffered K pipeline (copies never touch VGPRs)
    fetch_async(0, 0);
    wait_async0();
    __syncthreads();
#pragma unroll 1
    for (int kc = 0; kc < KCHUNKS; ++kc) {
        const int cur = kc & 1;
        const bool more = (kc + 1) < KCHUNKS;
        if (more) fetch_async(kc + 1, cur ^ 1);   // in flight during WMMA
        compute(cur);
        if (more) wait_async0();
        __syncthreads();
    }

    // write back: C/D layout -> M = r + 8*hf, N = l16
#pragma unroll
    for (int mi = 0; mi < 2; ++mi) {
#pragma unroll
        for (int ni = 0; ni < 4; ++ni) {
            const int col = obase + wn * 64 + ni * 16 + l16;
#pragma unroll
            for (int r = 0; r < 8; ++r) {
                const int lrow = wm * 32 + mi * 16 + r + 8 * hf;
                if (lrow < rows)
                    out[(size_t)(row_start + lrow) * OUT_FEAT + col] = acc[mi][ni][r];
            }
        }
    }
}

extern "C" void kernel_launch(void* const* d_in, const int* in_sizes, int n_in,
                              void* d_out, int out_size, void* d_ws, size_t ws_size,
                              hipStream_t stream) {
    const float* inp    = (const float*)d_in[0];
    const float* weight = (const float*)d_in[1];
    const int*   counts = (const int*)d_in[2];   // JAX default int32
    float* out = (float*)d_out;
    int*   ws  = (int*)d_ws;

    fmoe_build_tiles<<<dim3(1), dim3(1), 0, stream>>>(counts, ws);

    dim3 grid(MAX_TILES, OUT_FEAT / TN);   // fixed, deterministic grid
    dim3 block(256);
    fmoe_linear_wmma<<<grid, block, 0, stream>>>(inp, weight, out, ws);
}